// TimeEncoder_70755291234326
// MI455X (gfx1250) — compile-verified
//
#include <hip/hip_runtime.h>

#define N_TIME  100
#define OUT_DIM 8
#define BLOCK   256

typedef float v4f __attribute__((ext_vector_type(4)));
typedef int   v4i __attribute__((ext_vector_type(4)));

// Async global->LDS builtin wants (global int4*, shared int4*, imm offset, imm cpol).
typedef __attribute__((address_space(1))) v4i* g_v4i_ptr;
typedef __attribute__((address_space(3))) v4i* l_v4i_ptr;
typedef __attribute__((address_space(1))) int* g_i32_ptr;
typedef __attribute__((address_space(3))) int* l_i32_ptr;

__global__ __launch_bounds__(BLOCK) void time_encoder_kernel(
    const float* __restrict__ ts,     // [B, L+1]
    const float* __restrict__ W,      // [OUT_DIM, N_TIME] row-major
    const float* __restrict__ bias,   // [OUT_DIM]
    float* __restrict__ emb,          // [B*L, OUT_DIM]
    float* __restrict__ ts_out,       // [B*L]
    int L)
{
    // Staging buffer for raw W (async DMA target) + bias-folded transposed LUT.
    __shared__ __align__(16) float Wsh[OUT_DIM * N_TIME];   // 3.2 KB
    __shared__ __align__(16) float tbl[N_TIME][OUT_DIM];    // 3.2 KB, tbl[k][o] = W[o][k] + b[o]
    __shared__ __align__(16) float bsh[OUT_DIM];

    const int tid = threadIdx.x;

    // ---- Stage W into LDS via CDNA5 async global->LDS DMA (ASYNCcnt path) ----
#if __has_builtin(__builtin_amdgcn_global_load_async_to_lds_b128)
    for (int i = tid; i < (OUT_DIM * N_TIME) / 4; i += BLOCK) {
        __builtin_amdgcn_global_load_async_to_lds_b128(
            (g_v4i_ptr)(W + 4 * i), (l_v4i_ptr)(Wsh + 4 * i), 0, 0);
    }
  #if __has_builtin(__builtin_amdgcn_s_wait_asynccnt)
    __builtin_amdgcn_s_wait_asynccnt(0);
  #else
    asm volatile("s_wait_asynccnt 0" ::: "memory");
  #endif
#elif __has_builtin(__builtin_amdgcn_global_load_async_to_lds_b32)
    for (int i = tid; i < OUT_DIM * N_TIME; i += BLOCK) {
        __builtin_amdgcn_global_load_async_to_lds_b32(
            (g_i32_ptr)(W + i), (l_i32_ptr)(Wsh + i), 0, 0);
    }
  #if __has_builtin(__builtin_amdgcn_s_wait_asynccnt)
    __builtin_amdgcn_s_wait_asynccnt(0);
  #else
    asm volatile("s_wait_asynccnt 0" ::: "memory");
  #endif
#else
    for (int i = tid; i < OUT_DIM * N_TIME; i += BLOCK) Wsh[i] = W[i];
#endif
    if (tid < OUT_DIM) bsh[tid] = bias[tid];
    __syncthreads();

    // ---- Build transposed, bias-folded lookup table ----
    for (int i = tid; i < N_TIME * OUT_DIM; i += BLOCK) {
        const int k = i >> 3;          // time bucket
        const int o = i & 7;           // output channel
        tbl[k][o] = Wsh[o * N_TIME + k] + bsh[o];
    }
    __syncthreads();

    // ---- One (b, l) position per thread: gather + NT vector store ----
    const int l  = blockIdx.x * BLOCK + tid;
    const int bi = blockIdx.y;
    if (l >= L) return;

    const long long p   = (long long)bi * L + l;
    const float* row    = ts + (long long)bi * (L + 1) + l;
    const float  t0     = row[0];
    const float  t1     = row[1];
    const float  dt     = t1 - t0;              // PER_TIME == 1.0
    int idx = (int)floorf(dt);
    if (idx < 0) idx = 0;                       // relu(floor(.))

    v4f lo, hi;
    if (idx < N_TIME) {
        const v4f* t = (const v4f*)&tbl[idx][0];   // ds_load_b128 x2
        lo = t[0]; hi = t[1];
    } else {                                       // OOB one-hot -> bias only
        const v4f* t = (const v4f*)&bsh[0];
        lo = t[0]; hi = t[1];
    }

    // Streaming, write-once output: non-temporal global_store_b128 x2 + b32.
    v4f* o = (v4f*)(emb + p * OUT_DIM);
#if __has_builtin(__builtin_nontemporal_store)
    __builtin_nontemporal_store(lo, o);
    __builtin_nontemporal_store(hi, o + 1);
    __builtin_nontemporal_store(t0, ts_out + p);
#else
    o[0] = lo;
    o[1] = hi;
    ts_out[p] = t0;
#endif
}

extern "C" void kernel_launch(void* const* d_in, const int* in_sizes, int n_in,
                              void* d_out, int out_size, void* d_ws, size_t ws_size,
                              hipStream_t stream) {
    // setup_inputs order: input(int64, unused), timestamp(f32), W(f32), b(f32), train(int, unused)
    const float* ts   = (const float*)d_in[1];
    const float* W    = (const float*)d_in[2];
    const float* bias = (const float*)d_in[3];

    const int BL = in_sizes[0];               // B*L
    const int B  = in_sizes[1] - in_sizes[0]; // B*(L+1) - B*L
    const int L  = BL / B;

    float* emb    = (float*)d_out;                        // [B*L, 8]
    float* ts_out = (float*)d_out + (size_t)BL * OUT_DIM; // [B*L]

    dim3 block(BLOCK);
    dim3 grid((L + BLOCK - 1) / BLOCK, B);
    hipLaunchKernelGGL(time_encoder_kernel, grid, block, 0, stream,
                       ts, W, bias, emb, ts_out, L);
}